// RelationalGNNLayer_61134564491374
// MI455X (gfx1250) — compile-verified
//
#include <hip/hip_runtime.h>

typedef __attribute__((ext_vector_type(2))) float v2f;
typedef __attribute__((ext_vector_type(8))) float v8f;

// Padded LDS strides (floats) to avoid bank conflicts:
//  - s_mj stride 132: (8*132)%64 = 32, so lane L and L+16 hit different banks
//  - s_adj stride 66: consecutive rows land on different banks for A-frag loads
#define MJ_STRIDE  132
#define ADJ_STRIDE  66

// B=32, K=64, DIN=512, DOUT=512
// out[b,i,o] = deg[i]*(mi[b,i,o] + bias[o]) + sum_j adj[i,j]*mj[b,j,o]
// mi = x @ Wi^T, mj = x @ Wj^T  (Wi = W[:, :512], Wj = W[:, 512:])
__launch_bounds__(256)
__global__ void gnn_fused_kernel(const float* __restrict__ x,    // (32,64,512)
                                 const int*   __restrict__ adj,  // (64,64)
                                 const float* __restrict__ W,    // (512,1024)
                                 const float* __restrict__ bias, // (512,)
                                 float*       __restrict__ out)  // (32,64,512)
{
    __shared__ __align__(16) float s_mj [64 * MJ_STRIDE];   // mj tile: 64 rows x 128 cols
    __shared__ __align__(16) float s_adj[64 * ADJ_STRIDE];  // adj as float
    __shared__ float s_deg[64];

    const int tid  = threadIdx.x;
    const int wave = tid >> 5;       // 0..7
    const int lane = tid & 31;
    const int hf   = lane >> 4;      // lane half: 0/1
    const int l16  = lane & 15;

    const int b     = blockIdx.y;          // batch
    const int obase = blockIdx.x * 128;    // output-column tile base

    const int kslab = wave >> 1;           // 0..3 : 16-row slab in k-dim
    const int osub0 = (wave & 1) * 4;      // first of 4 owned 16-col subtiles

    // ---- stage adj -> LDS (as float), then deg[i] = sum_j adj[i][j] ----
    #pragma unroll
    for (int it = 0; it < 16; ++it) {               // 16*256 = 4096 = 64*64
        int idx = it * 256 + tid;
        int r = idx >> 6, c = idx & 63;
        s_adj[r * ADJ_STRIDE + c] = (float)adj[idx];
    }
    __syncthreads();
    if (tid < 64) {
        float d = 0.f;
        for (int j = 0; j < 64; ++j) d += s_adj[tid * ADJ_STRIDE + j];
        s_deg[tid] = d;
    }

    // ---- GEMM: mi, mj via V_WMMA_F32_16X16X4_F32, Kred = 512 ----
    // A frag (16x4 f32): a.x = A[M=l16][K=2*hf], a.y = A[M=l16][K=2*hf+1]
    // B frag (4x16 f32): b.x = B[K=2*hf][N=l16], b.y = B[K=2*hf+1][N=l16]
    const v8f zero = {0.f,0.f,0.f,0.f,0.f,0.f,0.f,0.f};
    v8f acc_i[4], acc_j[4];
    #pragma unroll
    for (int t = 0; t < 4; ++t) { acc_i[t] = zero; acc_j[t] = zero; }

    const float* xrow = x + ((size_t)b * 64 + kslab * 16 + l16) * 512 + 2 * hf;
    const float* wbase = W + (size_t)(obase + osub0 * 16 + l16) * 1024 + 2 * hf;

    for (int d = 0; d < 512; d += 4) {
        v2f a = *(const v2f*)(xrow + d);
        #pragma unroll
        for (int t = 0; t < 4; ++t) {
            const float* wrow = wbase + (size_t)t * 16 * 1024 + d;
            v2f bi = *(const v2f*)(wrow);         // Wi = W[:, :512]
            v2f bj = *(const v2f*)(wrow + 512);   // Wj = W[:, 512:]
            acc_i[t] = __builtin_amdgcn_wmma_f32_16x16x4_f32(
                false, a, false, bi, (short)0, acc_i[t], false, false);
            acc_j[t] = __builtin_amdgcn_wmma_f32_16x16x4_f32(
                false, a, false, bj, (short)0, acc_j[t], false, false);
        }
    }

    // ---- spill mj accumulators to LDS (C/D layout: VGPR r -> row r + 8*hf) ----
    #pragma unroll
    for (int t = 0; t < 4; ++t) {
        int col = (osub0 + t) * 16 + l16;         // 0..127 within block tile
        #pragma unroll
        for (int r = 0; r < 8; ++r) {
            int row = kslab * 16 + r + 8 * hf;
            s_mj[row * MJ_STRIDE + col] = acc_j[t][r];
        }
    }
    __syncthreads();

    // ---- aggregation: agg = adj(64x64) x mj(64x128), also WMMA f32 ----
    v8f agg[4];
    #pragma unroll
    for (int t = 0; t < 4; ++t) agg[t] = zero;

    const float* arow = s_adj + (kslab * 16 + l16) * ADJ_STRIDE + 2 * hf;
    for (int j = 0; j < 64; j += 4) {
        v2f a = *(const v2f*)(arow + j);          // 8B-aligned: ADJ_STRIDE even
        #pragma unroll
        for (int t = 0; t < 4; ++t) {
            int col = (osub0 + t) * 16 + l16;
            v2f bv;
            bv.x = s_mj[(j + 2 * hf)     * MJ_STRIDE + col];
            bv.y = s_mj[(j + 2 * hf + 1) * MJ_STRIDE + col];
            agg[t] = __builtin_amdgcn_wmma_f32_16x16x4_f32(
                false, a, false, bv, (short)0, agg[t], false, false);
        }
    }

    // ---- combine: out = deg[i]*(mi + bias) + agg ----
    float* outb = out + (size_t)b * 64 * 512;
    #pragma unroll
    for (int t = 0; t < 4; ++t) {
        int ocol = obase + (osub0 + t) * 16 + l16;
        float bv = bias[ocol];
        #pragma unroll
        for (int r = 0; r < 8; ++r) {
            int row = kslab * 16 + r + 8 * hf;
            outb[(size_t)row * 512 + ocol] = s_deg[row] * (acc_i[t][r] + bv) + agg[t][r];
        }
    }
}

extern "C" void kernel_launch(void* const* d_in, const int* in_sizes, int n_in,
                              void* d_out, int out_size, void* d_ws, size_t ws_size,
                              hipStream_t stream) {
    const float* x    = (const float*)d_in[0];
    const int*   adj  = (const int*)  d_in[1];
    const float* W    = (const float*)d_in[2];
    const float* bias = (const float*)d_in[3];
    float*       out  = (float*)d_out;

    dim3 grid(4, 32);   // 4 o-tiles of 128 cols x 32 batches
    gnn_fused_kernel<<<grid, 256, 0, stream>>>(x, adj, W, bias, out);
}